// DEMA_40707700032405
// MI455X (gfx1250) — compile-verified
//
#include <hip/hip_runtime.h>
#include <hip/hip_bf16.h>

typedef __attribute__((ext_vector_type(2))) float v2f;
typedef __attribute__((ext_vector_type(8))) float v8f;

// ---------------- compile-time coefficient tables ----------------
// DEMA recurrence:  z_t = M z_{t-1} + v x_t,  s_t = e1^T z_t
//   M = [[1-a, 1-a], [-a*b, 1-a*b]],  v = [a, a*b]
// Chunked (16-step) formulation:
//   S = Tlow * X + Acar * z_carry          (generic chunk)
//   S = T0   * X                           (chunk 0; init condition baked in)
// with Tlow[i][j] = e1^T M^(i-j) v  (j<=i),  Acar[i] = e1^T M^(i+1).

#define DEMA_A 0.3f
#define DEMA_B 0.1f

struct DTab {
  float TL[16][16];   // generic lower-triangular Toeplitz
  float T0[16][16];   // first chunk (s0 = x0, b0 = x1-x0 baked in)
  float AC[16][2];    // carry matrix rows: e1^T M^(i+1)
  float PW[17];       // (1-beta)^k
  float RINV;         // 1/(1-beta)
};

static constexpr DTab make_tab() {
  DTab d{};
  const float a = DEMA_A, b = DEMA_B;
  float cc[16] = {};
  {
    float u0 = a, u1 = a * b;          // u = M^k v, capture row-0
    for (int k = 0; k < 16; ++k) {
      cc[k] = u0;
      float n0 = (1.f - a) * (u0 + u1);
      float n1 = -a * b * u0 + (1.f - a * b) * u1;
      u0 = n0; u1 = n1;
    }
  }
  float r0[17] = {}, r1[17] = {};      // r[t] = e1^T M^t
  r0[0] = 1.f; r1[0] = 0.f;
  for (int t = 0; t < 16; ++t) {
    float n0 = r0[t] * (1.f - a) + r1[t] * (-a * b);
    float n1 = r0[t] * (1.f - a) + r1[t] * (1.f - a * b);
    r0[t + 1] = n0; r1[t + 1] = n1;
  }
  for (int i = 0; i < 16; ++i)
    for (int j = 0; j < 16; ++j)
      d.TL[i][j] = (j <= i) ? cc[i - j] : 0.f;
  for (int i = 0; i < 16; ++i) {
    for (int j = 0; j < 16; ++j) {
      float v;
      if (i == 0)       v = (j == 0) ? 1.f : 0.f;       // s_0 = x_0
      else if (j == 0)  v = r0[i] - r1[i];              // z0 = [x0; x1-x0]
      else if (j == 1)  v = r1[i] + cc[i - 1];
      else              v = (j <= i) ? cc[i - j] : 0.f;
      d.T0[i][j] = v;
    }
  }
  for (int i = 0; i < 16; ++i) { d.AC[i][0] = r0[i + 1]; d.AC[i][1] = r1[i + 1]; }
  {
    float p = 1.f;
    for (int k = 0; k <= 16; ++k) { d.PW[k] = p; p *= (1.f - b); }
  }
  d.RINV = 1.f / (1.f - b);
  return d;
}

__constant__ const DTab g_tab = make_tab();
static constexpr DTab h_tab = make_tab();   // compile-time folded copies

// SWAPX16: exchange lanes n <-> n+16 within the wave32.
// ds_swizzle group-of-32 encoding: xor=0x10, or=0, and=0x1f -> 0x401F.
__device__ __forceinline__ float swapx16(float v) {
  return __builtin_bit_cast(
      float, __builtin_amdgcn_ds_swizzle(__builtin_bit_cast(int, v), 0x401f));
}

// ---------------- kernel ----------------
// One wave (32 lanes) per block; block owns (batch, 16-channel tile).
// grid = 32 batches * 32 tiles = 1024 blocks.
__global__ __launch_bounds__(32) void dema_wmma_kernel(const float* __restrict__ x,
                                                       float* __restrict__ out) {
  constexpr int T = 4096, C = 512;
  const int tile  = blockIdx.x & 31;
  const int batch = blockIdx.x >> 5;
  const int lane  = threadIdx.x;          // wave32
  const int m     = lane & 15;            // row / channel-in-tile index
  const bool lo   = lane < 16;
  const int base  = lo ? 0 : 2;           // K-row offset for A/B VGPR layout
  const int ch    = tile * 16 + m;

  // Per-lane constant A-matrix registers (16x4 f32 A layout:
  // VGPR0 = K0(lanes0-15)/K2(lanes16-31), VGPR1 = K1/K3).
  v2f aT[4], aT0[4], acar;
#pragma unroll
  for (int kk = 0; kk < 4; ++kk) {
    const int j0 = 4 * kk + base;
    aT[kk].x  = g_tab.TL[m][j0];
    aT[kk].y  = g_tab.TL[m][j0 + 1];
    aT0[kk].x = g_tab.T0[m][j0];
    aT0[kk].y = g_tab.T0[m][j0 + 1];
  }
  acar.x = lo ? g_tab.AC[m][0] : 0.f;     // K=2,3 columns are zero padding
  acar.y = lo ? g_tab.AC[m][1] : 0.f;

  float wred[8];
#pragma unroll
  for (int r = 0; r < 8; ++r) wred[r] = lo ? h_tab.PW[15 - r] : h_tab.PW[7 - r];
  constexpr float pw16 = h_tab.PW[16];
  constexpr float rinv = h_tab.RINV;

  const size_t xbase = (size_t)batch * T * C + ch;

  // carry init: s_{-1}=x0 (so chunk-0 reduction's k=0 diff vanishes),
  // b_carry = b0/(1-beta) so (1-b)^16*b_carry = (1-b)^15*b0.
  const float x0 = __builtin_nontemporal_load(&x[xbase]);
  const float x1 = __builtin_nontemporal_load(&x[xbase + C]);
  float s_carry = x0;
  float b_carry = (x1 - x0) * rinv;       // compile-time reciprocal, no v_div_*

  // X chunk loads in B-matrix layout: VGPR0 = rows {4kk+base}, VGPR1 = {4kk+base+1}
  auto load_chunk = [&](int t0, float xv[8]) {
#pragma unroll
    for (int kk = 0; kk < 4; ++kk) {
      const int row = t0 + 4 * kk + base;
      xv[2 * kk]     = __builtin_nontemporal_load(&x[xbase + (size_t)row * C]);
      xv[2 * kk + 1] = __builtin_nontemporal_load(&x[xbase + (size_t)(row + 1) * C]);
    }
  };

  auto compute_chunk = [&](const v2f A[4], bool carry, const float xv[8], int t0) {
    v8f acc;
#pragma unroll
    for (int r = 0; r < 8; ++r) acc[r] = 0.f;
    if (carry) {
      v2f bc;                                   // B = [s; b; 0; 0] (4x16)
      bc.x = lo ? s_carry : 0.f;
      bc.y = lo ? b_carry : 0.f;
      acc = __builtin_amdgcn_wmma_f32_16x16x4_f32(false, acar, false, bc,
                                                  (short)0, acc, false, false);
    }
#pragma unroll
    for (int kk = 0; kk < 4; ++kk) {
      v2f bx;
      bx.x = xv[2 * kk];
      bx.y = xv[2 * kk + 1];
      acc = __builtin_amdgcn_wmma_f32_16x16x4_f32(false, A[kk], false, bx,
                                                  (short)0, acc, false, false);
    }
    // store D (streaming, non-temporal): VGPR r = rows r / r+8, N = lane&15
    const size_t obase = ((size_t)batch * T + t0 + (lo ? 0 : 8)) * C + ch;
#pragma unroll
    for (int r = 0; r < 8; ++r)
      __builtin_nontemporal_store(acc[r], &out[obase + (size_t)r * C]);

    // carry reconstruction from S:
    //   b15 = (1-b)^16 * b_prev + beta * sum_k (1-b)^(15-k) (s_k - s_{k-1})
    const float c87x = swapx16(acc[7]);         // lower<->upper: row7 <-> row15
    float prev = lo ? s_carry : c87x;
    float P = 0.f;
#pragma unroll
    for (int r = 0; r < 8; ++r) { P += wred[r] * (acc[r] - prev); prev = acc[r]; }
    P += swapx16(P);
    const float b15 = pw16 * b_carry + DEMA_B * P;
    const float s15 = lo ? c87x : (float)acc[7];
    s_carry = s15;
    b_carry = b15;
  };

  float xv[8], xn[8];
  load_chunk(0, xv);
  load_chunk(16, xn);                 // prefetch chunk 1
  compute_chunk(aT0, false, xv, 0);   // peeled chunk 0 (init condition baked in)

  for (int c = 1; c < 256; ++c) {
#pragma unroll
    for (int i = 0; i < 8; ++i) xv[i] = xn[i];
    if (c < 255) load_chunk(16 * (c + 1), xn);   // prefetch overlaps WMMAs
    compute_chunk(aT, true, xv, 16 * c);
  }
}

// ---------------- launch ----------------
extern "C" void kernel_launch(void* const* d_in, const int* in_sizes, int n_in,
                              void* d_out, int out_size, void* d_ws, size_t ws_size,
                              hipStream_t stream) {
  const float* x = (const float*)d_in[0];
  float* out = (float*)d_out;
  (void)in_sizes; (void)n_in; (void)out_size; (void)d_ws; (void)ws_size;
  dim3 grid(32 * 32);   // 32 batches x 32 channel tiles
  dim3 block(32);       // one wave32 per block
  hipLaunchKernelGGL(dema_wmma_kernel, grid, block, 0, stream, x, out);
}